// LINE2nd_Model_33973191311965
// MI455X (gfx1250) — compile-verified
//
#include <hip/hip_runtime.h>

typedef __attribute__((ext_vector_type(2))) float v2f;
typedef __attribute__((ext_vector_type(8))) float v8f;

// Numerically stable log-sigmoid using hw transcendentals.
__device__ __forceinline__ float log_sigmoid(float x) {
    float ax = __builtin_fabsf(x);
    return __builtin_fminf(x, 0.f) - __logf(1.f + __expf(-ax));
}

__global__ void zero_out_kernel(float* out) {
    if (threadIdx.x == 0 && blockIdx.x == 0) *out = 0.f;
}

// One wave32 handles 16 batch elements. For each of the (1+K) "other" matrices
// it computes a 16x16 Gram tile with V_WMMA_F32_16X16X4_F32 accumulated over
// d=128 in 32 chunks of K=4, then extracts the diagonal (the needed dots).
__global__ void __launch_bounds__(256)
line2nd_wmma_kernel(const int* __restrict__ pos_v,
                    const int* __restrict__ pos_u,
                    const int* __restrict__ neg_v,
                    const float* __restrict__ weights,
                    const float* __restrict__ emb,
                    float* __restrict__ out,
                    int K, int n_tiles)
{
    const int lane = threadIdx.x & 31;
    const int wave = threadIdx.x >> 5;
    const int tile = blockIdx.x * (blockDim.x >> 5) + wave;
    if (tile >= n_tiles) return;           // wave-uniform exit: EXEC stays all-ones

    const int b0   = tile * 16;
    const int rsel = lane & 15;            // matrix row this lane serves
    const int hi   = lane >> 4;            // 0: K=0..1 of chunk, 1: K=2..3
    const int b    = b0 + rsel;

    // ---- Load A = emb_v tile (16x128 f32) in WMMA 16x4-chunk A-layout ----
    const int av = pos_v[b];
    const float* arow = emb + (size_t)av * 128 + (hi << 1);
    v2f aF[32];
#pragma unroll
    for (int c = 0; c < 32; ++c)
        aF[c] = *(const v2f*)(arow + c * 4);

    float acc = 0.f;
    for (int m = 0; m <= K; ++m) {
        const int br = (m == 0) ? pos_u[b] : neg_v[b * K + (m - 1)];
        const float* brow = emb + (size_t)br * 128 + (hi << 1);

        v8f C = {};
#pragma unroll
        for (int c = 0; c < 32; ++c) {
            v2f bF = *(const v2f*)(brow + c * 4);
            // 8 args: (neg_a, A, neg_b, B, c_mod, C, reuse_a, reuse_b)
            C = __builtin_amdgcn_wmma_f32_16x16x4_f32(
                    false, aF[c], false, bF, (short)0, C, false, false);
        }

        // Diagonal D[i][i]: lanes 0-7 hold i=lane in C[i];
        // lanes 24-31 hold i=lane-16 in C[lane-24]. Other lanes: don't care.
        const int r = (lane < 8) ? lane : (lane - 24);
        float d = (r == 0) ? C[0] : (r == 1) ? C[1] : (r == 2) ? C[2] :
                  (r == 3) ? C[3] : (r == 4) ? C[4] : (r == 5) ? C[5] :
                  (r == 6) ? C[6] : C[7];

        acc += (m == 0) ? log_sigmoid(d) : log_sigmoid(-d);
    }

    // Weighted per-row contribution; only diagonal-holding lanes participate.
    const bool participant = (lane < 8) || (lane >= 24);
    const int  row = (lane < 8) ? lane : (lane - 16);
    float v = participant ? (weights[b0 + row] * acc) : 0.f;

    // wave32 butterfly reduction
#pragma unroll
    for (int off = 16; off > 0; off >>= 1)
        v += __shfl_xor(v, off, 32);

    if (lane == 0)
        atomicAdd(out, -v);                // loss = -sum(w * score)
}

extern "C" void kernel_launch(void* const* d_in, const int* in_sizes, int n_in,
                              void* d_out, int out_size, void* d_ws, size_t ws_size,
                              hipStream_t stream) {
    const int*   pos_v   = (const int*)d_in[0];
    const int*   pos_u   = (const int*)d_in[1];
    const int*   neg_v   = (const int*)d_in[2];
    const float* weights = (const float*)d_in[3];
    const float* emb     = (const float*)d_in[4];
    float*       out     = (float*)d_out;

    const int Bsz = in_sizes[0];
    const int K   = in_sizes[2] / Bsz;          // 10
    const int n_tiles = (Bsz + 15) / 16;        // 1024

    zero_out_kernel<<<1, 32, 0, stream>>>(out);

    const int waves_per_block = 8;              // 256 threads
    const int grid = (n_tiles + waves_per_block - 1) / waves_per_block;
    line2nd_wmma_kernel<<<grid, waves_per_block * 32, 0, stream>>>(
        pos_v, pos_u, neg_v, weights, emb, out, K, n_tiles);
}